// GeodesicSpectralModel_42296837931419
// MI455X (gfx1250) — compile-verified
//
#include <hip/hip_runtime.h>

typedef __attribute__((ext_vector_type(2))) float v2f;
typedef __attribute__((ext_vector_type(8))) float v8f;

#define N_PTS 11
#define N_NEWTON 5
#define METRIC_FLOOR 0.1f

// lane <-> lane^16 full-wave32 swap (ds_swizzle group-of-32: xor=0x10, and=0x1f)
__device__ inline float swap16f(float x) {
    return __int_as_float(__builtin_amdgcn_ds_swizzle(__float_as_int(x), 0x401F));
}

__device__ inline float ftanh(float x) {
    float e = __expf(2.0f * x);
    return 1.0f - 2.0f / (e + 1.0f);
}

__global__ __launch_bounds__(128) void geodesic_kernel(
    const float* __restrict__ c_src, const float* __restrict__ c_tgt,
    const float* __restrict__ wl,
    const float* __restrict__ W1g, const float* __restrict__ b1g,
    const float* __restrict__ W2g, const float* __restrict__ b2g,
    const float* __restrict__ W3g, const float* __restrict__ b3g,
    const float* __restrict__ D1g, const float* __restrict__ d1g,
    const float* __restrict__ D2g, const float* __restrict__ d2g,
    float* __restrict__ out, int nelem)
{
    // Stage decoder weights in LDS (used once per element at the end).
    __shared__ float sD1[6 * 64];
    __shared__ float sd1[64];
    __shared__ float sD2[64];
    for (int i = threadIdx.x; i < 6 * 64; i += blockDim.x) sD1[i] = D1g[i];
    for (int i = threadIdx.x; i < 64; i += blockDim.x) { sd1[i] = d1g[i]; sD2[i] = D2g[i]; }
    __syncthreads();

    const int lane = threadIdx.x & 31;
    const int half = lane >> 4;          // 0: rows 0-7, 1: rows 8-15
    const bool lo  = (half == 0);
    const int gw   = (blockIdx.x * blockDim.x + threadIdx.x) >> 5; // global wave id
    const int elem = gw * 16 + (lane & 15);
    const int elemc = (elem < nelem) ? elem : (nelem - 1);

    // ---- per-lane constant weights (registers) ----
    float W1c[8], W1l[8], b1v[8], b2v[8], W3v[8];
#pragma unroll
    for (int r = 0; r < 8; ++r) {
        int m = r + 8 * half;
        W1c[r] = W1g[m];          // W1[0, m]
        W1l[r] = W1g[16 + m];     // W1[1, m]
        b1v[r] = b1g[m];
        b2v[r] = b2g[m];
        W3v[r] = W3g[m];
    }
    // A = W2^T in WMMA A-layout, 4 K-chunks of 2 VGPRs. A[m][k] = W2[k][m].
    // lanes 0-15: K = 4kc+0/1 ; lanes 16-31: K = 4kc+2/3 ; M = lane%16.
    const int mm = lane & 15;
    v2f Aw[4];
#pragma unroll
    for (int kc = 0; kc < 4; ++kc) {
        int k0 = 4 * kc + 2 * half;
        Aw[kc].x = W2g[k0 * 16 + mm];
        Aw[kc].y = W2g[(k0 + 1) * 16 + mm];
    }
    const float b3s = b3g[0];
    const float d2s = d2g[0];

    const float lam = wl[elemc];
    const float c0  = c_src[elemc];
    const float c1  = c_tgt[elemc];

    // D-layout activations (v8f per lane) -> B-layout chunks, then 4 chained
    // f32 16x16x4 WMMAs:  D(16x16) = W2^T(16x16) * H^T(16x16)
    auto mmB = [&](v8f h) -> v8f {
        float s0 = swap16f(h[0]), s1 = swap16f(h[1]);
        float s2 = swap16f(h[2]), s3 = swap16f(h[3]);
        float s4 = swap16f(h[4]), s5 = swap16f(h[5]);
        float s6 = swap16f(h[6]), s7 = swap16f(h[7]);
        v2f B0 = { lo ? h[0] : s2,  lo ? h[1] : s3 };   // K = 0..3
        v2f B1 = { lo ? h[4] : s6,  lo ? h[5] : s7 };   // K = 4..7
        v2f B2 = { lo ? s0 : h[2],  lo ? s1 : h[3] };   // K = 8..11
        v2f B3 = { lo ? s4 : h[6],  lo ? s5 : h[7] };   // K = 12..15
        v8f acc = {};
        acc = __builtin_amdgcn_wmma_f32_16x16x4_f32(false, Aw[0], false, B0, (short)0, acc, false, false);
        acc = __builtin_amdgcn_wmma_f32_16x16x4_f32(false, Aw[1], false, B1, (short)0, acc, false, false);
        acc = __builtin_amdgcn_wmma_f32_16x16x4_f32(false, Aw[2], false, B2, (short)0, acc, false, false);
        acc = __builtin_amdgcn_wmma_f32_16x16x4_f32(false, Aw[3], false, B3, (short)0, acc, false, false);
        return acc;
    };

    // metric with value + d/dc + d2/dc2  (Newton phase)
    auto metric3 = [&](float c, float& g, float& gp, float& gpp) {
        v8f h = {}, hp = {}, hq = {};
#pragma unroll
        for (int r = 0; r < 8; ++r) {
            float a  = fmaf(W1c[r], c, fmaf(W1l[r], lam, b1v[r]));
            float ap = W1c[r];
            float t  = ftanh(a);
            float sq = 1.0f - t * t;
            float tp = sq * ap;
            h[r]  = t;
            hp[r] = tp;
            hq[r] = -2.0f * t * tp * ap;
        }
        v8f pv = mmB(h);
        v8f pp = mmB(hp);
        v8f pq = mmB(hq);
        float dv = 0.f, dp = 0.f, dq = 0.f;
#pragma unroll
        for (int r = 0; r < 8; ++r) {
            float a  = pv[r] + b2v[r];
            float t  = ftanh(a);
            float sq = 1.0f - t * t;
            float tp = sq * pp[r];
            float tq = fmaf(-2.0f * t * tp, pp[r], sq * pq[r]);
            dv = fmaf(t,  W3v[r], dv);
            dp = fmaf(tp, W3v[r], dp);
            dq = fmaf(tq, W3v[r], dq);
        }
        dv += swap16f(dv); dp += swap16f(dp); dq += swap16f(dq);
        float s = dv + b3s;
        float e = __expf(-fabsf(s));
        float l = __logf(1.0f + e);
        g = fmaxf(s, 0.0f) + l + METRIC_FLOOR;
        float sig = (s >= 0.0f ? 1.0f : e) / (1.0f + e);
        gp  = sig * dp;
        gpp = fmaf(sig * (1.0f - sig) * dp, dp, sig * dq);
    };

    // metric with value + d/dc only (final trajectory phase)
    auto metric2 = [&](float c, float& g, float& gp) {
        v8f h = {}, hp = {};
#pragma unroll
        for (int r = 0; r < 8; ++r) {
            float a  = fmaf(W1c[r], c, fmaf(W1l[r], lam, b1v[r]));
            float t  = ftanh(a);
            float sq = 1.0f - t * t;
            h[r]  = t;
            hp[r] = sq * W1c[r];
        }
        v8f pv = mmB(h);
        v8f pp = mmB(hp);
        float dv = 0.f, dp = 0.f;
#pragma unroll
        for (int r = 0; r < 8; ++r) {
            float a  = pv[r] + b2v[r];
            float t  = ftanh(a);
            float tp = (1.0f - t * t) * pp[r];
            dv = fmaf(t,  W3v[r], dv);
            dp = fmaf(tp, W3v[r], dp);
        }
        dv += swap16f(dv); dp += swap16f(dp);
        float s = dv + b3s;
        float e = __expf(-fabsf(s));
        g = fmaxf(s, 0.0f) + __logf(1.0f + e) + METRIC_FLOOR;
        float sig = (s >= 0.0f ? 1.0f : e) / (1.0f + e);
        gp = sig * dp;
    };

    auto christo3 = [&](float c, float& G, float& dG) {
        float g, gp, gpp; metric3(c, g, gp, gpp);
        float ig = 1.0f / g;
        G  = 0.5f * gp * ig;
        dG = 0.5f * (gpp - gp * gp * ig) * ig;
    };
    auto christo2 = [&](float c) -> float {
        float g, gp; metric2(c, g, gp);
        return 0.5f * gp / g;
    };

    const float dt = 1.0f / (float)(N_PTS - 1);

    // ---- Newton shooting on v0 (tangent seeded ct=0, vt=1) ----
    float v0 = c1 - c0;
    for (int it = 0; it < N_NEWTON; ++it) {
        float c = c0, v = v0, ct = 0.0f, vt = 1.0f;
        for (int s = 0; s < N_PTS - 1; ++s) {
            auto der = [&](float cc, float vv, float cct, float vvt,
                           float& dc, float& dv, float& dct, float& dvt) {
                float G, dG; christo3(cc, G, dG);
                dc  = vv;
                dv  = -G * vv * vv;
                dct = vvt;
                dvt = -(dG * cct * vv * vv + 2.0f * G * vv * vvt);
            };
            float k1c,k1v,k1ct,k1vt, k2c,k2v,k2ct,k2vt;
            float k3c,k3v,k3ct,k3vt, k4c,k4v,k4ct,k4vt;
            der(c, v, ct, vt, k1c,k1v,k1ct,k1vt);
            der(c+0.5f*dt*k1c, v+0.5f*dt*k1v, ct+0.5f*dt*k1ct, vt+0.5f*dt*k1vt,
                k2c,k2v,k2ct,k2vt);
            der(c+0.5f*dt*k2c, v+0.5f*dt*k2v, ct+0.5f*dt*k2ct, vt+0.5f*dt*k2vt,
                k3c,k3v,k3ct,k3vt);
            der(c+dt*k3c, v+dt*k3v, ct+dt*k3ct, vt+dt*k3vt,
                k4c,k4v,k4ct,k4vt);
            const float w = dt / 6.0f;
            c  += w * (k1c  + 2.0f*k2c  + 2.0f*k3c  + k4c);
            v  += w * (k1v  + 2.0f*k2v  + 2.0f*k3v  + k4v);
            ct += w * (k1ct + 2.0f*k2ct + 2.0f*k3ct + k4ct);
            vt += w * (k1vt + 2.0f*k2vt + 2.0f*k3vt + k4vt);
        }
        float dr  = ct;
        float drs = (fabsf(dr) < 1e-6f) ? 1e-6f : dr;
        v0 = v0 - (c - c1) / drs;
    }

    // ---- final integration + trajectory statistics ----
    float c = c0, v = v0;
    float sum_c = c, sum_v = v, maxav = fabsf(v), plen = 0.0f, cprev = c;
    for (int s = 0; s < N_PTS - 1; ++s) {
        auto der2 = [&](float cc, float vv, float& dc, float& dv) {
            float G = christo2(cc);
            dc = vv;
            dv = -G * vv * vv;
        };
        float k1c,k1v,k2c,k2v,k3c,k3v,k4c,k4v;
        der2(c, v, k1c, k1v);
        der2(c + 0.5f*dt*k1c, v + 0.5f*dt*k1v, k2c, k2v);
        der2(c + 0.5f*dt*k2c, v + 0.5f*dt*k2v, k3c, k3v);
        der2(c + dt*k3c, v + dt*k3v, k4c, k4v);
        const float w = dt / 6.0f;
        c += w * (k1c + 2.0f*k2c + 2.0f*k3c + k4c);
        v += w * (k1v + 2.0f*k2v + 2.0f*k3v + k4v);
        sum_c += c;
        sum_v += v;
        maxav  = fmaxf(maxav, fabsf(v));
        plen  += fabsf(c - cprev);
        cprev  = c;
    }

    float feats[6];
    feats[0] = sum_c * (1.0f / (float)N_PTS);
    feats[1] = c;
    feats[2] = plen;
    feats[3] = sum_v * (1.0f / (float)N_PTS);
    feats[4] = maxav;
    feats[5] = lam;

    // ---- decoder MLP 6 -> 64 -> 1 ----
    float acc = d2s;
    for (int j = 0; j < 64; ++j) {
        float a = sd1[j];
#pragma unroll
        for (int i = 0; i < 6; ++i) a = fmaf(feats[i], sD1[i * 64 + j], a);
        acc = fmaf(ftanh(a), sD2[j], acc);
    }

    if (lo && elem < nelem) out[elem] = acc;
}

extern "C" void kernel_launch(void* const* d_in, const int* in_sizes, int n_in,
                              void* d_out, int out_size, void* d_ws, size_t ws_size,
                              hipStream_t stream) {
    const float* c_src = (const float*)d_in[0];
    const float* c_tgt = (const float*)d_in[1];
    const float* wl    = (const float*)d_in[2];
    const float* W1    = (const float*)d_in[3];
    const float* b1    = (const float*)d_in[4];
    const float* W2    = (const float*)d_in[5];
    const float* b2    = (const float*)d_in[6];
    const float* W3    = (const float*)d_in[7];
    const float* b3    = (const float*)d_in[8];
    const float* D1    = (const float*)d_in[9];
    const float* d1    = (const float*)d_in[10];
    const float* D2    = (const float*)d_in[11];
    const float* d2    = (const float*)d_in[12];
    float* out = (float*)d_out;

    int n = in_sizes[0];
    // 16 elements per wave32, 4 waves (128 threads) per block -> 64 elems/block
    int blocks = (n + 63) / 64;
    hipLaunchKernelGGL(geodesic_kernel, dim3(blocks), dim3(128), 0, stream,
                       c_src, c_tgt, wl, W1, b1, W2, b2, W3, b3,
                       D1, d1, D2, d2, out, n);
}